// MaskRCNN_35201551958787
// MI455X (gfx1250) — compile-verified
//
#include <hip/hip_runtime.h>
#include <hip/hip_bf16.h>
#include <cstdint>

// ---------------------------------------------------------------------------
// Mask R-CNN post-processing for MI455X (gfx1250).
// Outputs (flat, in reference return order):
//   [0,400)            boxes   (100,4)
//   [400,500)          scores_top (100)
//   [500,600)          labels  (100)  (stored as float)
//   [600,600+64e6)     masks   (100,1,800,800)
// ---------------------------------------------------------------------------

#define N_DET       100
#define NUM_CLASSES 81
#define MASK_RES    28
#define OUT_RES     800
#define ROWS_PER_BLK 16          // rows of the 800x800 output per block
#define BLK_THREADS 256          // 8 waves (wave32)

// gfx1250 async global->LDS path, guarded so host pass / older toolchains
// fall back to a plain copy (compile always succeeds; histogram tells truth).
#if defined(__AMDGCN__) && __has_builtin(__builtin_amdgcn_global_load_async_to_lds_b128)
#define USE_ASYNC_LDS 1
#endif

static __device__ __forceinline__ int iclamp(int v, int lo, int hi) {
    return v < lo ? lo : (v > hi ? hi : v);
}

typedef float vfloat4 __attribute__((ext_vector_type(4)));
typedef int   vint4   __attribute__((ext_vector_type(4)));

#if USE_ASYNC_LDS
typedef __attribute__((address_space(1))) vint4 as1_vint4;   // global
typedef __attribute__((address_space(3))) vint4 as3_vint4;   // LDS
#endif

// ---------------------------------------------------------------------------
// Kernel A: softmax-top / argmax / box decode.  100 rows -> one tiny block.
// ---------------------------------------------------------------------------
__global__ void postproc_kernel(const float* __restrict__ proposals,
                                const float* __restrict__ scores,
                                const float* __restrict__ deltas,
                                const int*   __restrict__ Hp,
                                const int*   __restrict__ Wp,
                                float* __restrict__ out,
                                int*   __restrict__ labels_ws) {
    int i = blockIdx.x * blockDim.x + threadIdx.x;
    if (i >= N_DET) return;

    // --- softmax top prob + argmax (first-max semantics like jnp.argmax) ---
    const float* s = scores + (size_t)i * NUM_CLASSES;
    float m = s[0];
    int lab = 0;
    for (int j = 1; j < NUM_CLASSES; ++j) {
        float v = s[j];
        if (v > m) { m = v; lab = j; }
    }
    float sum = 0.0f;
    for (int j = 0; j < NUM_CLASSES; ++j) sum += expf(s[j] - m);
    float top = 1.0f / sum;   // == exp(m) / sum(exp(s))

    // --- box decode: apply first 4 deltas, clip to [0,W]x[0,H] ---
    const float* p = proposals + (size_t)i * 4;
    const float* d = deltas    + (size_t)i * (4 * NUM_CLASSES);
    float w  = p[2] - p[0];
    float h  = p[3] - p[1];
    float cx = p[0] + 0.5f * w;
    float cy = p[1] + 0.5f * h;
    float pcx = d[0] * w + cx;
    float pcy = d[1] * h + cy;
    float pw  = expf(d[2]) * w;
    float ph  = expf(d[3]) * h;
    float Wf = (float)Wp[0];
    float Hf = (float)Hp[0];
    float x1 = fminf(fmaxf(pcx - 0.5f * pw, 0.0f), Wf);
    float y1 = fminf(fmaxf(pcy - 0.5f * ph, 0.0f), Hf);
    float x2 = fminf(fmaxf(pcx + 0.5f * pw, 0.0f), Wf);
    float y2 = fminf(fmaxf(pcy + 0.5f * ph, 0.0f), Hf);

    out[i * 4 + 0] = x1;
    out[i * 4 + 1] = y1;
    out[i * 4 + 2] = x2;
    out[i * 4 + 3] = y2;
    out[N_DET * 4 + i]         = top;
    out[N_DET * 4 + N_DET + i] = (float)lab;
    labels_ws[i] = lab;
}

// ---------------------------------------------------------------------------
// Kernel B: bilinear 28x28 -> 800x800 resize of the selected class mask.
// grid = (800/ROWS_PER_BLK, N_DET); block = 256 threads.
// Stage the 3136-byte tile into LDS with GLOBAL_LOAD_ASYNC_TO_LDS_B128,
// row-interpolate once into an LDS line buffer, then stream NT float4 stores.
// ---------------------------------------------------------------------------
__global__ void resize_masks_kernel(const float* __restrict__ mask_logits,
                                    const int*   __restrict__ labels,
                                    float* __restrict__ masks_out) {
    const int det = blockIdx.y;
    const int r0  = blockIdx.x * ROWS_PER_BLK;
    const int t   = threadIdx.x;

    __shared__ __align__(16) float tile[MASK_RES * MASK_RES];      // 784 f32
    __shared__ float line[ROWS_PER_BLK * MASK_RES];                // 448 f32

    const int lab = labels[det];
    const float* src = mask_logits +
        ((size_t)det * NUM_CLASSES + (size_t)lab) * (MASK_RES * MASK_RES);

#if USE_ASYNC_LDS
    // 784 floats = 196 x b128 transfers; one per lane for t < 196.
    if (t < (MASK_RES * MASK_RES) / 4) {
        __builtin_amdgcn_global_load_async_to_lds_b128(
            (as1_vint4*)(src + t * 4),
            (as3_vint4*)(tile + t * 4),
            0, 0);
    }
#if __has_builtin(__builtin_amdgcn_s_wait_asynccnt)
    __builtin_amdgcn_s_wait_asynccnt(0);
#else
    asm volatile("s_wait_asynccnt 0" ::: "memory");
#endif
    __syncthreads();
#else
    for (int i = t; i < MASK_RES * MASK_RES; i += BLK_THREADS) tile[i] = src[i];
    __syncthreads();
#endif

    // Row interpolation: line[r][c] = lerp over y for output row (r0+r).
    // Half-pixel convention: fy = (y+0.5)*(28/800) - 0.5, clamp indices
    // (index clamping == JAX's linear-kernel weight renormalization at edges).
    const float sc = (float)MASK_RES / (float)OUT_RES;   // 0.035 exactly
    for (int i = t; i < ROWS_PER_BLK * MASK_RES; i += BLK_THREADS) {
        int r = i / MASK_RES;
        int c = i - r * MASK_RES;
        float fy  = ((float)(r0 + r) + 0.5f) * sc - 0.5f;
        float fiy = floorf(fy);
        float ty  = fy - fiy;
        int   iy  = (int)fiy;
        int   y0  = iclamp(iy,     0, MASK_RES - 1);
        int   y1  = iclamp(iy + 1, 0, MASK_RES - 1);
        float a = tile[y0 * MASK_RES + c];
        float b = tile[y1 * MASK_RES + c];
        line[i] = a + ty * (b - a);
    }
    __syncthreads();

    // Column interpolation + streaming NT stores, 4 px (16 B) per thread/iter.
    float* orow = masks_out + (size_t)det * (OUT_RES * OUT_RES)
                            + (size_t)r0 * OUT_RES;
    const int nvec = ROWS_PER_BLK * (OUT_RES / 4);       // 3200 float4s
    for (int v = t; v < nvec; v += BLK_THREADS) {
        int r  = v / (OUT_RES / 4);
        int c4 = (v - r * (OUT_RES / 4)) * 4;
        const float* lp = &line[r * MASK_RES];
        vfloat4 val;
#pragma unroll
        for (int k = 0; k < 4; ++k) {
            float fx  = ((float)(c4 + k) + 0.5f) * sc - 0.5f;
            float fix = floorf(fx);
            float tx  = fx - fix;
            int   ixi = (int)fix;
            int   x0  = iclamp(ixi,     0, MASK_RES - 1);
            int   x1  = iclamp(ixi + 1, 0, MASK_RES - 1);
            float a = lp[x0];
            float b = lp[x1];
            val[k] = a + tx * (b - a);
        }
        __builtin_nontemporal_store(
            val, (vfloat4*)(orow + (size_t)r * OUT_RES + c4));
    }
}

// ---------------------------------------------------------------------------
extern "C" void kernel_launch(void* const* d_in, const int* in_sizes, int n_in,
                              void* d_out, int out_size, void* d_ws, size_t ws_size,
                              hipStream_t stream) {
    const float* proposals   = (const float*)d_in[0];
    const float* scores      = (const float*)d_in[1];
    const float* bbox_deltas = (const float*)d_in[2];
    const float* mask_logits = (const float*)d_in[3];
    const int*   Hp          = (const int*)d_in[4];
    const int*   Wp          = (const int*)d_in[5];

    float* out       = (float*)d_out;
    int*   labels_ws = (int*)d_ws;     // 100 ints of scratch

    postproc_kernel<<<1, 128, 0, stream>>>(proposals, scores, bbox_deltas,
                                           Hp, Wp, out, labels_ws);

    dim3 grid(OUT_RES / ROWS_PER_BLK, N_DET);   // (50, 100)
    resize_masks_kernel<<<grid, BLK_THREADS, 0, stream>>>(
        mask_logits, labels_ws, out + (N_DET * 4 + N_DET + N_DET));
}